// PPOCoLightNetwork_1443109012157
// MI455X (gfx1250) — compile-verified
//
#include <hip/hip_runtime.h>
#include <hip/hip_bf16.h>

#define BB    8
#define AA    1024
#define BA    (BB * AA)     // 8192 agents
#define NN    5
#define DIN   32
#define HIDN  128
#define HEADS 5
#define HDIM  32
#define PPH   8             // num phases P
#define PD    64
#define GOUT  160           // HDIM*HEADS

typedef __attribute__((ext_vector_type(2))) float v2f;
typedef __attribute__((ext_vector_type(8))) float v8f;

// V_WMMA_F32_16X16X4_F32 : D = A(16x4) x B(4x16) + C, full fp32.
static __device__ __forceinline__ v8f wmma4(v2f a, v2f b, v8f c) {
  return __builtin_amdgcn_wmma_f32_16x16x4_f32(
      /*neg_a=*/false, a, /*neg_b=*/false, b,
      /*c_mod=*/(short)0, c, /*reuse_a=*/false, /*reuse_b=*/false);
}

static __device__ __forceinline__ float wave_sum(float x) {
#pragma unroll
  for (int off = 16; off > 0; off >>= 1) x += __shfl_xor(x, off, 32);
  return x;
}
static __device__ __forceinline__ int wave_min(int x) {
#pragma unroll
  for (int off = 16; off > 0; off >>= 1) {
    int o = __shfl_xor(x, off, 32);
    x = (o < x) ? o : x;
  }
  return x;
}

// ---------------------------------------------------------------- embed MLP
// block = 256 threads (8 waves) handles 16 agents; two fused GEMM+ReLU+LN.
__global__ __launch_bounds__(256) void k_embed(
    const float* __restrict__ obs,
    const float* __restrict__ w0, const float* __restrict__ b0,
    const float* __restrict__ g0, const float* __restrict__ be0,
    const float* __restrict__ w1, const float* __restrict__ b1,
    const float* __restrict__ g1, const float* __restrict__ be1,
    float* __restrict__ hout) {
  __shared__ float obs_s[16][DIN];
  __shared__ float h0s[16][HIDN];
  __shared__ float h1s[16][HIDN];
  __shared__ float red[16][16], red2[16][16];
  __shared__ float mu[16], rv[16];

  const int tid = threadIdx.x;
  const int wid = tid >> 5, lane = tid & 31;
  const int half = lane >> 4, lr = lane & 15;
  const int t16 = blockIdx.x * 16;

  for (int i = tid; i < 16 * DIN; i += 256)
    obs_s[i >> 5][i & 31] = obs[(size_t)(t16 + (i >> 5)) * DIN + (i & 31)];
  __syncthreads();

  // GEMM0: [16,32] @ [32,128]; wave wid owns 16-col tile wid
  {
    const int nc0 = wid * 16;
    v8f acc = {};
#pragma unroll
    for (int k0 = 0; k0 < DIN; k0 += 4) {
      const int kk = k0 + 2 * half;
      v2f a; a.x = obs_s[lr][kk];            a.y = obs_s[lr][kk + 1];
      v2f b; b.x = w0[kk * HIDN + nc0 + lr]; b.y = w0[(kk + 1) * HIDN + nc0 + lr];
      acc = wmma4(a, b, acc);
    }
#pragma unroll
    for (int r = 0; r < 8; ++r) {
      const int m = r + 8 * half, c = nc0 + lr;
      const float v = acc[r] + b0[c];
      h0s[m][c] = v > 0.f ? v : 0.f;
    }
  }
  __syncthreads();

  // LayerNorm over 128 cols (16 rows x 16 partial threads)
  {
    const int row = tid >> 4, c0 = tid & 15;
    float s = 0.f, s2 = 0.f;
#pragma unroll
    for (int j = 0; j < 8; ++j) { const float x = h0s[row][c0 + 16 * j]; s += x; s2 += x * x; }
    red[row][c0] = s; red2[row][c0] = s2;
    __syncthreads();
    if (tid < 16) {
      float ss = 0.f, ss2 = 0.f;
      for (int j = 0; j < 16; ++j) { ss += red[tid][j]; ss2 += red2[tid][j]; }
      const float m = ss / 128.f;
      mu[tid] = m;
      rv[tid] = rsqrtf(ss2 / 128.f - m * m + 1e-5f);
    }
    __syncthreads();
#pragma unroll
    for (int j = 0; j < 8; ++j) {
      const int c = c0 + 16 * j;
      h0s[row][c] = (h0s[row][c] - mu[row]) * rv[row] * g0[c] + be0[c];
    }
  }
  __syncthreads();

  // GEMM1: [16,128] @ [128,128]
  {
    const int nc0 = wid * 16;
    v8f acc = {};
    for (int k0 = 0; k0 < HIDN; k0 += 4) {
      const int kk = k0 + 2 * half;
      v2f a; a.x = h0s[lr][kk];              a.y = h0s[lr][kk + 1];
      v2f b; b.x = w1[kk * HIDN + nc0 + lr]; b.y = w1[(kk + 1) * HIDN + nc0 + lr];
      acc = wmma4(a, b, acc);
    }
#pragma unroll
    for (int r = 0; r < 8; ++r) {
      const int m = r + 8 * half, c = nc0 + lr;
      const float v = acc[r] + b1[c];
      h1s[m][c] = v > 0.f ? v : 0.f;
    }
  }
  __syncthreads();

  // LayerNorm + write out
  {
    const int row = tid >> 4, c0 = tid & 15;
    float s = 0.f, s2 = 0.f;
#pragma unroll
    for (int j = 0; j < 8; ++j) { const float x = h1s[row][c0 + 16 * j]; s += x; s2 += x * x; }
    red[row][c0] = s; red2[row][c0] = s2;
    __syncthreads();
    if (tid < 16) {
      float ss = 0.f, ss2 = 0.f;
      for (int j = 0; j < 16; ++j) { ss += red[tid][j]; ss2 += red2[tid][j]; }
      const float m = ss / 128.f;
      mu[tid] = m;
      rv[tid] = rsqrtf(ss2 / 128.f - m * m + 1e-5f);
    }
    __syncthreads();
#pragma unroll
    for (int j = 0; j < 8; ++j) {
      const int c = c0 + 16 * j;
      hout[(size_t)(t16 + row) * HIDN + c] =
          (h1s[row][c] - mu[row]) * rv[row] * g1[c] + be1[c];
    }
  }
}

// ------------------------------------------------- one-hot adjacency -> nidx
// wave per agent; lanes scan 1024-wide one-hot rows.
__global__ __launch_bounds__(256) void k_nidx(const float* __restrict__ adj,
                                              int* __restrict__ nidx) {
  const int agent = blockIdx.x * 8 + (threadIdx.x >> 5);
  const int lane = threadIdx.x & 31;
  for (int n = 0; n < NN; ++n) {
    const size_t base = ((size_t)agent * NN + n) * AA;
    int found = AA;
    for (int j = lane; j < AA; j += 32)
      if (adj[base + j] > 0.5f) found = (j < found) ? j : found;
    found = wave_min(found);
    if (lane == 0) nidx[agent * NN + n] = (found < AA) ? found : 0;
  }
}

// ---------------------------------------------- q/k/v projections (128->160)
__global__ __launch_bounds__(256) void k_proj(
    const float* __restrict__ h,
    const float* __restrict__ wq, const float* __restrict__ bq,
    const float* __restrict__ wk, const float* __restrict__ bk,
    const float* __restrict__ wv, const float* __restrict__ bv,
    float* __restrict__ oq, float* __restrict__ ok, float* __restrict__ ov) {
  __shared__ float As[16][HIDN];
  const float* W; const float* bias; float* out;
  if (blockIdx.y == 0)      { W = wq; bias = bq; out = oq; }
  else if (blockIdx.y == 1) { W = wk; bias = bk; out = ok; }
  else                      { W = wv; bias = bv; out = ov; }

  const int tid = threadIdx.x, wid = tid >> 5, lane = tid & 31;
  const int half = lane >> 4, lr = lane & 15;
  const int t16 = blockIdx.x * 16;

  for (int i = tid; i < 16 * HIDN; i += 256)
    As[i >> 7][i & 127] = h[(size_t)(t16 + (i >> 7)) * HIDN + (i & 127)];
  __syncthreads();

  for (int nt = wid; nt < 10; nt += 8) {   // 160 = 10 col-tiles
    const int nc0 = nt * 16;
    v8f acc = {};
    for (int k0 = 0; k0 < HIDN; k0 += 4) {
      const int kk = k0 + 2 * half;
      v2f a; a.x = As[lr][kk];               a.y = As[lr][kk + 1];
      v2f b; b.x = W[kk * GOUT + nc0 + lr];  b.y = W[(kk + 1) * GOUT + nc0 + lr];
      acc = wmma4(a, b, acc);
    }
#pragma unroll
    for (int r = 0; r < 8; ++r) {
      const int m = r + 8 * half, c = nc0 + lr;
      out[(size_t)(t16 + m) * GOUT + c] = acc[r] + bias[c];
    }
  }
}

// --------------------------------------------------------- GAT attention
// wave per agent; lane = head-dim channel (32). Edge MLP in VALU.
__global__ __launch_bounds__(256) void k_gat(
    const float* __restrict__ hq, const float* __restrict__ hk,
    const float* __restrict__ hv, const int* __restrict__ nidx,
    const float* __restrict__ phase,
    const float* __restrict__ ew1, const float* __restrict__ eb1,
    const float* __restrict__ ew2, const float* __restrict__ eb2,
    float* __restrict__ agg) {
  const int agent = blockIdx.x * 8 + (threadIdx.x >> 5);
  const int lane = threadIdx.x & 31;
  const int b = agent >> 10, a = agent & 1023;

  int nrow[NN];
#pragma unroll
  for (int n = 0; n < NN; ++n) nrow[n] = b * AA + nidx[agent * NN + n];

  // edge-bias MLP: lane n (<5) computes 5 head biases for neighbor n
  float ebv[HEADS];
  {
    const int n = (lane < NN) ? lane : 0;
    const int j = nidx[agent * NN + n];
    const float er = (float)(a >> 5), ec = (float)(a & 31);
    const float nr = (float)(j >> 5), nc = (float)(j & 31);
    const float adr = fabsf(nr - er), adc = fabsf(nc - ec);
    const float man = adr + adc;
    const float dist = man / 62.f;
    const float nh = (adc >= adr) ? 1.f : 0.f;
    const float ph = phase[agent];
    const float same = (nh == ph) ? 1.f : 0.f;
    const float conn = (man > 0.f) ? 1.f : 0.f;
    float hid[16];
#pragma unroll
    for (int i = 0; i < 16; ++i) {
      const float s = eb1[i] + same * ew1[i] + (1.f - same) * ew1[16 + i] +
                      dist * ew1[32 + i] + conn * ew1[48 + i];
      hid[i] = s > 0.f ? s : 0.f;
    }
#pragma unroll
    for (int hh = 0; hh < HEADS; ++hh) {
      float e = eb2[hh];
#pragma unroll
      for (int i = 0; i < 16; ++i) e += hid[i] * ew2[i * HEADS + hh];
      ebv[hh] = e;
    }
  }

  float acc = 0.f;
  const size_t qb = (size_t)agent * GOUT;
#pragma unroll
  for (int hh = 0; hh < HEADS; ++hh) {
    const float qv = hq[qb + lane * HEADS + hh];
    float lg[NN];
#pragma unroll
    for (int n = 0; n < NN; ++n) {
      const float kv = hk[(size_t)nrow[n] * GOUT + lane * HEADS + hh];
      lg[n] = wave_sum(qv * kv) + __shfl(ebv[hh], n, 32);
    }
    float mx = lg[0];
#pragma unroll
    for (int n = 1; n < NN; ++n) mx = fmaxf(mx, lg[n]);
    float se = 0.f;
#pragma unroll
    for (int n = 0; n < NN; ++n) { lg[n] = __expf(lg[n] - mx); se += lg[n]; }
    const float inv = 1.f / se;
#pragma unroll
    for (int n = 0; n < NN; ++n)
      acc += lg[n] * inv * hv[(size_t)nrow[n] * GOUT + lane * HEADS + hh];
  }
  agg[(size_t)agent * HDIM + lane] = acc * 0.2f;  // mean over 5 heads
}

// ------------------------------------------------------- h2 = relu(agg@go_w)
__global__ __launch_bounds__(256) void k_h2(
    const float* __restrict__ agg, const float* __restrict__ gow,
    const float* __restrict__ gob, float* __restrict__ h2) {
  __shared__ float As[16][HDIM];
  const int tid = threadIdx.x, wid = tid >> 5, lane = tid & 31;
  const int half = lane >> 4, lr = lane & 15;
  const int t16 = blockIdx.x * 16;
  for (int i = tid; i < 16 * HDIM; i += 256)
    As[i >> 5][i & 31] = agg[(size_t)(t16 + (i >> 5)) * HDIM + (i & 31)];
  __syncthreads();
  const int nc0 = wid * 16;
  v8f acc = {};
#pragma unroll
  for (int k0 = 0; k0 < HDIM; k0 += 4) {
    const int kk = k0 + 2 * half;
    v2f a; a.x = As[lr][kk];                a.y = As[lr][kk + 1];
    v2f b; b.x = gow[kk * HIDN + nc0 + lr]; b.y = gow[(kk + 1) * HIDN + nc0 + lr];
    acc = wmma4(a, b, acc);
  }
#pragma unroll
  for (int r = 0; r < 8; ++r) {
    const int m = r + 8 * half, c = nc0 + lr;
    const float v = acc[r] + gob[c];
    h2[(size_t)(t16 + m) * HIDN + c] = v > 0.f ? v : 0.f;
  }
}

// ----------------------------------------------------- global context mean
__global__ __launch_bounds__(128) void k_gmean(const float* __restrict__ h2,
                                               float* __restrict__ g) {
  const int b = blockIdx.x, d = threadIdx.x;
  float s = 0.f;
  for (int a = 0; a < AA; ++a) s += h2[((size_t)b * AA + a) * HIDN + d];
  g[b * HIDN + d] = s / (float)AA;
}

// ------------------------------------------------------- phase actor head
// block handles 4 agents (2 pair-tiles of 16 rows = 2 agents x 8 phases).
__global__ __launch_bounds__(256) void k_phase(
    const float* __restrict__ h2,
    const float* __restrict__ pew, const float* __restrict__ peb,
    const float* __restrict__ peg, const float* __restrict__ pebe,
    const float* __restrict__ aqw, const float* __restrict__ aqb,
    const float* __restrict__ akw, const float* __restrict__ akb,
    const float* __restrict__ avw, const float* __restrict__ avb,
    const float* __restrict__ aow, const float* __restrict__ aob,
    const float* __restrict__ h1w, const float* __restrict__ h1b,
    const float* __restrict__ h2w, const float* __restrict__ h2b,
    float* __restrict__ logits) {
  __shared__ float h2s[4][HIDN];
  __shared__ float pes[32][PD];     // pe, later reused for ao@ao_w output
  __shared__ float qas[32][PD];
  __shared__ float kas[32][PD];
  __shared__ float vas[32][PD];
  __shared__ float aos[32][PD];
  __shared__ float h1s[32][32];
  __shared__ float red[32][8], red2[32][8];
  __shared__ float mu[32], rv[32];

  const int tid = threadIdx.x, wid = tid >> 5, lane = tid & 31;
  const int half = lane >> 4, lr = lane & 15;
  const int ag0 = blockIdx.x * 4;

  for (int i = tid; i < 4 * HIDN; i += 256)
    h2s[i >> 7][i & 127] = h2[(size_t)(ag0 + (i >> 7)) * HIDN + (i & 127)];
  __syncthreads();

  // pe = relu([h2 | eye] @ pe_w + pe_b) : K=136, wave = (pair, ntile)
  {
    const int pair = wid >> 2, nt = wid & 3, nc0 = nt * 16;
    const int agl = pair * 2 + (lr >> 3);   // local agent for A-row lr
    const int p = lr & 7;
    v8f acc = {};
    for (int k0 = 0; k0 < 136; k0 += 4) {
      const int kk = k0 + 2 * half;
      v2f a;
      a.x = (kk < 128)     ? h2s[agl][kk]     : ((kk - 128) == p ? 1.f : 0.f);
      a.y = (kk + 1 < 128) ? h2s[agl][kk + 1] : ((kk + 1 - 128) == p ? 1.f : 0.f);
      v2f b; b.x = pew[kk * PD + nc0 + lr]; b.y = pew[(kk + 1) * PD + nc0 + lr];
      acc = wmma4(a, b, acc);
    }
#pragma unroll
    for (int r = 0; r < 8; ++r) {
      const int m = r + 8 * half, c = nc0 + lr;
      const float v = acc[r] + peb[c];
      pes[pair * 16 + m][c] = v > 0.f ? v : 0.f;
    }
  }
  __syncthreads();

  // LayerNorm over 64 cols, 32 rows
  {
    const int row = tid >> 3, c0 = tid & 7;
    float s = 0.f, s2 = 0.f;
#pragma unroll
    for (int j = 0; j < 8; ++j) { const float x = pes[row][c0 + 8 * j]; s += x; s2 += x * x; }
    red[row][c0] = s; red2[row][c0] = s2;
    __syncthreads();
    if (tid < 32) {
      float ss = 0.f, ss2 = 0.f;
      for (int j = 0; j < 8; ++j) { ss += red[tid][j]; ss2 += red2[tid][j]; }
      const float m = ss / 64.f;
      mu[tid] = m;
      rv[tid] = rsqrtf(ss2 / 64.f - m * m + 1e-5f);
    }
    __syncthreads();
#pragma unroll
    for (int j = 0; j < 8; ++j) {
      const int c = c0 + 8 * j;
      pes[row][c] = (pes[row][c] - mu[row]) * rv[row] * peg[c] + pebe[c];
    }
  }
  __syncthreads();

  // qa / ka / va : [32,64] @ [64,64]
  for (int m3 = 0; m3 < 3; ++m3) {
    const float* W  = (m3 == 0) ? aqw : (m3 == 1) ? akw : avw;
    const float* bb = (m3 == 0) ? aqb : (m3 == 1) ? akb : avb;
    float (*dst)[PD] = (m3 == 0) ? qas : (m3 == 1) ? kas : vas;
    const int pair = wid >> 2, nt = wid & 3, nc0 = nt * 16;
    v8f acc = {};
    for (int k0 = 0; k0 < PD; k0 += 4) {
      const int kk = k0 + 2 * half;
      v2f a; a.x = pes[pair * 16 + lr][kk];  a.y = pes[pair * 16 + lr][kk + 1];
      v2f b; b.x = W[kk * PD + nc0 + lr];    b.y = W[(kk + 1) * PD + nc0 + lr];
      acc = wmma4(a, b, acc);
    }
#pragma unroll
    for (int r = 0; r < 8; ++r)
      dst[pair * 16 + r + 8 * half][nc0 + lr] = acc[r] + bb[nc0 + lr];
  }
  __syncthreads();

  // 4-head 8x8 self-attention per agent: thread = (agent, p, head)
  if (tid < 128) {
    const int agl = tid >> 5, rem = tid & 31, p = rem >> 2, hh = rem & 3;
    const int base = (agl >> 1) * 16 + (agl & 1) * 8;
    const int rowp = base + p;
    float lg[PPH];
#pragma unroll
    for (int q = 0; q < PPH; ++q) {
      float s = 0.f;
#pragma unroll
      for (int d = 0; d < 16; ++d) s += qas[rowp][hh * 16 + d] * kas[base + q][hh * 16 + d];
      lg[q] = s * 0.25f;  // 1/sqrt(16)
    }
    float mx = lg[0];
#pragma unroll
    for (int q = 1; q < PPH; ++q) mx = fmaxf(mx, lg[q]);
    float se = 0.f;
#pragma unroll
    for (int q = 0; q < PPH; ++q) { lg[q] = __expf(lg[q] - mx); se += lg[q]; }
    const float inv = 1.f / se;
#pragma unroll
    for (int d = 0; d < 16; ++d) {
      float s = 0.f;
#pragma unroll
      for (int q = 0; q < PPH; ++q) s += lg[q] * inv * vas[base + q][hh * 16 + d];
      aos[rowp][hh * 16 + d] = s;
    }
  }
  __syncthreads();

  // ao = aos @ ao_w + ao_b  (reuse pes as output)
  {
    const int pair = wid >> 2, nt = wid & 3, nc0 = nt * 16;
    v8f acc = {};
    for (int k0 = 0; k0 < PD; k0 += 4) {
      const int kk = k0 + 2 * half;
      v2f a; a.x = aos[pair * 16 + lr][kk];  a.y = aos[pair * 16 + lr][kk + 1];
      v2f b; b.x = aow[kk * PD + nc0 + lr];  b.y = aow[(kk + 1) * PD + nc0 + lr];
      acc = wmma4(a, b, acc);
    }
#pragma unroll
    for (int r = 0; r < 8; ++r)
      pes[pair * 16 + r + 8 * half][nc0 + lr] = acc[r] + aob[nc0 + lr];
  }
  __syncthreads();

  // h1 = relu(ao @ h1_w + h1_b) : [32,64]@[64,32] -> waves 0..3
  if (wid < 4) {
    const int pair = wid >> 1, nt = wid & 1, nc0 = nt * 16;
    v8f acc = {};
    for (int k0 = 0; k0 < PD; k0 += 4) {
      const int kk = k0 + 2 * half;
      v2f a; a.x = pes[pair * 16 + lr][kk]; a.y = pes[pair * 16 + lr][kk + 1];
      v2f b; b.x = h1w[kk * 32 + nc0 + lr]; b.y = h1w[(kk + 1) * 32 + nc0 + lr];
      acc = wmma4(a, b, acc);
    }
#pragma unroll
    for (int r = 0; r < 8; ++r) {
      const float v = acc[r] + h1b[nc0 + lr];
      h1s[pair * 16 + r + 8 * half][nc0 + lr] = v > 0.f ? v : 0.f;
    }
  }
  __syncthreads();

  // logits = h1 @ h2_w + h2_b
  if (tid < 32) {
    float s = h2b[0];
#pragma unroll
    for (int j = 0; j < 32; ++j) s += h1s[tid][j] * h2w[j];
    const int pair = tid >> 4, agin = (tid >> 3) & 1, p = tid & 7;
    logits[(size_t)(ag0 + pair * 2 + agin) * PPH + p] = s;
  }
}

// --------------------------------------------------------------- critic
__global__ __launch_bounds__(128) void k_critic(
    const float* __restrict__ h2, const float* __restrict__ g,
    const float* __restrict__ c1w, const float* __restrict__ c1b,
    const float* __restrict__ c2w, const float* __restrict__ c2b,
    const float* __restrict__ c3w, const float* __restrict__ c3b,
    float* __restrict__ val) {
  __shared__ float cs[16][256];
  __shared__ float c1s[16][64];
  __shared__ float c2s[16][32];
  const int tid = threadIdx.x, wid = tid >> 5, lane = tid & 31;
  const int half = lane >> 4, lr = lane & 15;
  const int t16 = blockIdx.x * 16;
  const int b = t16 >> 10;

  for (int i = tid; i < 16 * 256; i += 128) {
    const int row = i >> 8, c = i & 255;
    cs[row][c] = (c < 128) ? h2[(size_t)(t16 + row) * HIDN + c]
                           : g[b * HIDN + (c - 128)];
  }
  __syncthreads();

  {  // [16,256] @ [256,64] + relu
    const int nc0 = wid * 16;
    v8f acc = {};
    for (int k0 = 0; k0 < 256; k0 += 4) {
      const int kk = k0 + 2 * half;
      v2f a; a.x = cs[lr][kk];              a.y = cs[lr][kk + 1];
      v2f b; b.x = c1w[kk * 64 + nc0 + lr]; b.y = c1w[(kk + 1) * 64 + nc0 + lr];
      acc = wmma4(a, b, acc);
    }
#pragma unroll
    for (int r = 0; r < 8; ++r) {
      const int m = r + 8 * half, c = nc0 + lr;
      const float v = acc[r] + c1b[c];
      c1s[m][c] = v > 0.f ? v : 0.f;
    }
  }
  __syncthreads();

  if (wid < 2) {  // [16,64] @ [64,32] + relu
    const int nc0 = wid * 16;
    v8f acc = {};
    for (int k0 = 0; k0 < 64; k0 += 4) {
      const int kk = k0 + 2 * half;
      v2f a; a.x = c1s[lr][kk];             a.y = c1s[lr][kk + 1];
      v2f b; b.x = c2w[kk * 32 + nc0 + lr]; b.y = c2w[(kk + 1) * 32 + nc0 + lr];
      acc = wmma4(a, b, acc);
    }
#pragma unroll
    for (int r = 0; r < 8; ++r) {
      const int m = r + 8 * half, c = nc0 + lr;
      const float v = acc[r] + c2b[c];
      c2s[m][c] = v > 0.f ? v : 0.f;
    }
  }
  __syncthreads();

  if (tid < 16) {
    float s = c3b[0];
#pragma unroll
    for (int j = 0; j < 32; ++j) s += c2s[tid][j] * c3w[j];
    val[t16 + tid] = s;
  }
}

// ----------------------------------------------------------------- launch
extern "C" void kernel_launch(void* const* d_in, const int* in_sizes, int n_in,
                              void* d_out, int out_size, void* d_ws, size_t ws_size,
                              hipStream_t stream) {
  (void)in_sizes; (void)n_in; (void)out_size; (void)ws_size;
  int i = 0;
  const float* obs    = (const float*)d_in[i++];
  const float* adj    = (const float*)d_in[i++];
  const float* phase  = (const float*)d_in[i++];
  const float* emb_w0 = (const float*)d_in[i++]; const float* emb_b0 = (const float*)d_in[i++];
  const float* emb_g0 = (const float*)d_in[i++]; const float* emb_be0= (const float*)d_in[i++];
  const float* emb_w1 = (const float*)d_in[i++]; const float* emb_b1 = (const float*)d_in[i++];
  const float* emb_g1 = (const float*)d_in[i++]; const float* emb_be1= (const float*)d_in[i++];
  const float* gq_w = (const float*)d_in[i++]; const float* gq_b = (const float*)d_in[i++];
  const float* gk_w = (const float*)d_in[i++]; const float* gk_b = (const float*)d_in[i++];
  const float* gv_w = (const float*)d_in[i++]; const float* gv_b = (const float*)d_in[i++];
  const float* go_w = (const float*)d_in[i++]; const float* go_b = (const float*)d_in[i++];
  const float* ew1  = (const float*)d_in[i++]; const float* eb1  = (const float*)d_in[i++];
  const float* ew2  = (const float*)d_in[i++]; const float* eb2  = (const float*)d_in[i++];
  const float* pe_w = (const float*)d_in[i++]; const float* pe_b = (const float*)d_in[i++];
  const float* pe_g = (const float*)d_in[i++]; const float* pe_be= (const float*)d_in[i++];
  const float* aq_w = (const float*)d_in[i++]; const float* aq_b = (const float*)d_in[i++];
  const float* ak_w = (const float*)d_in[i++]; const float* ak_b = (const float*)d_in[i++];
  const float* av_w = (const float*)d_in[i++]; const float* av_b = (const float*)d_in[i++];
  const float* ao_w = (const float*)d_in[i++]; const float* ao_b = (const float*)d_in[i++];
  const float* h1_w = (const float*)d_in[i++]; const float* h1_b = (const float*)d_in[i++];
  const float* h2_w = (const float*)d_in[i++]; const float* h2_b = (const float*)d_in[i++];
  const float* c1_w = (const float*)d_in[i++]; const float* c1_b = (const float*)d_in[i++];
  const float* c2_w = (const float*)d_in[i++]; const float* c2_b = (const float*)d_in[i++];
  const float* c3_w = (const float*)d_in[i++]; const float* c3_b = (const float*)d_in[i++];

  float* ws  = (float*)d_ws;
  float* h   = ws;                               // [BA,128]
  float* hq  = h  + (size_t)BA * HIDN;           // [BA,160]
  float* hk  = hq + (size_t)BA * GOUT;           // [BA,160]
  float* hv  = hk + (size_t)BA * GOUT;           // [BA,160]
  float* agg = hv + (size_t)BA * GOUT;           // [BA,32]
  float* h2  = agg + (size_t)BA * HDIM;          // [BA,128]
  float* g   = h2 + (size_t)BA * HIDN;           // [B,128]
  int* nidx  = (int*)(g + BB * HIDN);            // [BA,5]

  float* out_logits = (float*)d_out;             // [B,A,P]
  float* out_value  = out_logits + (size_t)BA * PPH;  // [B,A]

  k_embed<<<BA / 16, 256, 0, stream>>>(obs, emb_w0, emb_b0, emb_g0, emb_be0,
                                       emb_w1, emb_b1, emb_g1, emb_be1, h);
  k_nidx<<<BA / 8, 256, 0, stream>>>(adj, nidx);
  k_proj<<<dim3(BA / 16, 3), 256, 0, stream>>>(h, gq_w, gq_b, gk_w, gk_b,
                                               gv_w, gv_b, hq, hk, hv);
  k_gat<<<BA / 8, 256, 0, stream>>>(hq, hk, hv, nidx, phase, ew1, eb1, ew2, eb2, agg);
  k_h2<<<BA / 16, 256, 0, stream>>>(agg, go_w, go_b, h2);
  k_gmean<<<BB, 128, 0, stream>>>(h2, g);
  k_phase<<<BA / 4, 256, 0, stream>>>(h2, pe_w, pe_b, pe_g, pe_be,
                                      aq_w, aq_b, ak_w, ak_b, av_w, av_b,
                                      ao_w, ao_b, h1_w, h1_b, h2_w, h2_b,
                                      out_logits);
  k_critic<<<BA / 16, 128, 0, stream>>>(h2, g, c1_w, c1_b, c2_w, c2_b,
                                        c3_w, c3_b, out_value);
}